// CSB_60043642798884
// MI455X (gfx1250) — compile-verified
//
#include <hip/hip_runtime.h>

typedef __attribute__((ext_vector_type(2))) float v2f;
typedef __attribute__((ext_vector_type(4))) float v4f;
typedef __attribute__((ext_vector_type(8))) float v8f;
typedef unsigned int u32x4 __attribute__((ext_vector_type(4)));
typedef int          i32x4 __attribute__((ext_vector_type(4)));
typedef int          i32x8 __attribute__((ext_vector_type(8)));

#define C_IN  64
#define HFULL 256
#define WFULL 256
#define HH    128
#define WW    128
#define MK    256      // M = K = 4*C
#define KC    32       // K chunk staged in LDS
#define SWS   36       // sW row stride in floats (32 data + 4 TDM pad): conflict-free A b64 loads
#define PCOL  144      // padded cols of pair-interleaved F panel (288 floats/row ≡ 32 banks)

__global__ __launch_bounds__(512)
void quad_mix_wmma_kernel(const float* __restrict__ x,
                          const float* __restrict__ w,
                          float* __restrict__ out)
{
    __shared__ float sW[MK * SWS];          // 256 x 36 x 4B = 36 KB  (W K-panel, TDM-filled)
    __shared__ v2f   sFp[(KC / 2) * PCOL];  // 16 x 144 x 8B = 18 KB  (F panel, K-pair interleaved)
    // element (k, col) lives at sFp[(k>>1)*PCOL + col][k&1]

    const int tid  = threadIdx.x;
    const int wave = tid >> 5;       // 0..15  -> M rows [16*wave, 16*wave+16)
    const int lane = tid & 31;
    const int l16  = lane & 15;
    const int hi   = lane >> 4;      // 0: lanes 0-15, 1: lanes 16-31
    const int koff = hi << 1;        // A/B operand K offset per ISA layout (0 or 2)

    const int bid = blockIdx.x;      // 2048 blocks: one (b, hh) slab each
    const int bb  = bid >> 7;        // batch 0..15
    const int hh  = bid & 127;       // row in half space

    v8f acc[8];
#pragma unroll
    for (int t = 0; t < 8; ++t) acc[t] = (v8f)0.0f;

    const unsigned ldsW = (unsigned)(uintptr_t)(void*)sW;   // low 32 bits = LDS byte offset

    for (int chunk = 0; chunk < MK / KC; ++chunk) {
        const int kc0 = chunk * KC;
        __syncthreads();

        // ---- stage W[:, kc0:kc0+32] into LDS via Tensor Data Mover ----------
        // 2D tile: 32 contiguous floats (x) by 256 rows (y, stride 256 floats),
        // LDS pad 4 DWORDs after every 32 DWORDs -> padded row stride SWS=36.
        if (wave == 0) {
            const unsigned long long ga =
                (unsigned long long)(uintptr_t)(w + kc0);   // tile start, bytes
            u32x4 g0;
            g0[0] = 1u;                                     // count=1, user descriptor
            g0[1] = ldsW;                                   // lds_addr
            g0[2] = (unsigned)(ga & 0xFFFFFFFFu);           // global_addr[31:0]
            g0[3] = (unsigned)((ga >> 32) & 0x01FFFFFFu)    // global_addr[56:32]
                  | (2u << 30);                             // type = 2 ("image")
            i32x8 g1;
            g1[0] = (2 << 16)       // data_size = 4 bytes
                  | (1 << 20)       // pad_enable
                  | (4 << 22)       // pad_interval: every 32 DWORDs (one row)
                  | (3 << 25);      // pad_amount: 4 DWORDs
            g1[1] = (32  << 16);    // tensor_dim0[15:0]  = 32  (bits 79:48)
            g1[2] = (256 << 16);    // tensor_dim1[15:0]  = 256 (bits 111:80)
            g1[3] = (32  << 16);    // tile_dim0 = 32 (bits 127:112)
            g1[4] = 256;            // tile_dim1 = 256; tile_dim2 = 0
            g1[5] = 256;            // tensor_dim0_stride[31:0] = 256 floats
            g1[6] = 0;              // stride high bits / tensor_dim1_stride lo
            g1[7] = 0;
            i32x4 gz = (i32x4)0;
#if __clang_major__ >= 23
            i32x8 gz8 = (i32x8)0;
            __builtin_amdgcn_tensor_load_to_lds(g0, g1, gz, gz, gz8, 0);
#else
            __builtin_amdgcn_tensor_load_to_lds(g0, g1, gz, gz, 0);
#endif
        }

        // ---- stage F chunk, K-pair interleaved (VGPR path) ------------------
        // wave w owns pair-row w: f_all channels c = kc0+2w and c+1 (same quadrant)
        {
            const int c  = kc0 + 2 * wave;   // even channel
            const int q  = c >> 6;           // quadrant: 0 TL, 1 BL, 2 TR, 3 BR
            const int c0 = c & 63;
            const float* gx0 = x
                + (size_t)(((bb * C_IN + c0) * HFULL) + (hh + HH * (q & 1))) * WFULL
                + (WW * (q >> 1));
            const float* gx1 = gx0 + (size_t)HFULL * WFULL;   // channel c+1
            v2f* dst = &sFp[wave * PCOL];
#pragma unroll
            for (int i = 0; i < 4; ++i) {
                const int col = lane + 32 * i;     // coalesced 128B per b32 load
                v2f p;
                p.x = gx0[col];
                p.y = gx1[col];
                dst[col] = p;                      // ds_store_b64, conflict-free
            }
            if (chunk + 1 < MK / KC) {             // warm next chunk (global_prefetch_b8)
                const int cn  = kc0 + KC + 2 * wave;
                const int qn  = cn >> 6;
                const int c0n = cn & 63;
                const float* gn0 = x
                    + (size_t)(((bb * C_IN + c0n) * HFULL) + (hh + HH * (qn & 1))) * WFULL
                    + (WW * (qn >> 1));
                __builtin_prefetch(gn0 + 4 * lane, 0, 1);
                __builtin_prefetch(gn0 + (size_t)HFULL * WFULL + 4 * lane, 0, 1);
            }
        }

        if (wave == 0) {
            __builtin_amdgcn_s_wait_tensorcnt(0);  // W panel landed in LDS
        }
        __syncthreads();

        // ---- compute: 8 k-steps of 4, 8 N-tiles of 16 -----------------------
        const int rowA = (wave << 4) + l16;
#pragma unroll
        for (int kt = 0; kt < KC / 4; ++kt) {
            const int ka = 4 * kt + koff;
            // A 16x4 fp32: lanes0-15 K={4kt,4kt+1}, lanes16-31 K={4kt+2,4kt+3}
            v2f a = *(const v2f*)(&sW[rowA * SWS + ka]);
            // B pair-row: lanes0-15 read {k,k+1}, lanes16-31 {k+2,k+3} -> one b64 each
            const v2f* brow = &sFp[(2 * kt + hi) * PCOL + l16];
#pragma unroll
            for (int t = 0; t < 8; ++t) {
                acc[t] = __builtin_amdgcn_wmma_f32_16x16x4_f32(
                    false, a, false, brow[t << 4], (short)0, acc[t], false, false);
            }
        }
    }

    // ---- epilogue: scatter D back into quadrant layout ----------------------
    // D layout: VGPR v -> M = v (lanes 0-15) / M = 8+v (lanes 16-31), N = lane15
#pragma unroll
    for (int v = 0; v < 8; ++v) {
        const int o  = (wave << 4) + (hi << 3) + v;   // output channel 0..255
        const int qg = o >> 6;
        const int o0 = o & 63;
        float* go = out
            + (size_t)(((bb * C_IN + o0) * HFULL) + (hh + HH * (qg & 1))) * WFULL
            + (WW * (qg >> 1)) + l16;
#pragma unroll
        for (int t = 0; t < 8; ++t) {
            go[t << 4] = acc[t][v];
        }
    }
}

extern "C" void kernel_launch(void* const* d_in, const int* in_sizes, int n_in,
                              void* d_out, int out_size, void* d_ws, size_t ws_size,
                              hipStream_t stream) {
    const float* x = (const float*)d_in[0];   // [16, 64, 256, 256] fp32
    const float* w = (const float*)d_in[1];   // [256, 256] fp32
    float* out = (float*)d_out;               // [16, 64, 256, 256] fp32

    dim3 grid(16 * 128);   // one block per (batch, half-row)
    dim3 block(512);       // 16 waves (wave32)
    hipLaunchKernelGGL(quad_mix_wmma_kernel, grid, block, 0, stream, x, w, out);
}